// Net_14869176779172
// MI455X (gfx1250) — compile-verified
//
#include <hip/hip_runtime.h>
#include <cstdint>

// Problem constants (from reference)
#define NBATCH 1024
#define NDENDS 4096
#define NSYNS  60
#define CUT    1638            // int(4096 * 0.4)

// Tiling
#define BT     16              // batches per block tile (WMMA N)
#define DCH    8               // dendrites per double-buffered chunk (2 WMMA K-steps)
#define NCHUNK 8               // chunks per block -> 64 dendrites per block
#define DRANGE (DCH * NCHUNK)  // 64
#define ND_T   (NDENDS / DRANGE) // 64 dendrite ranges
#define NB_T   (NBATCH / BT)     // 64 batch tiles

#define TILE_ELEMS   (DCH * NSYNS)          // 480 f32 per row (1920 B)
#define ROW_STRIDE_E (NDENDS * NSYNS)       // 245760 f32 between batches (fits 32b)

typedef __attribute__((ext_vector_type(2))) float v2f;
typedef __attribute__((ext_vector_type(8))) float v8f;
typedef __attribute__((ext_vector_type(4))) unsigned int v4u;
typedef __attribute__((ext_vector_type(8))) int v8i;
typedef __attribute__((ext_vector_type(4))) int v4i;

// USE_WS=true : deterministic partials into workspace [NBATCH][ND_T]
// USE_WS=false: atomic fallback straight into out (pre-initialized with soma_b)
template <bool USE_WS>
__global__ __launch_bounds__(32) void dend_stream_kernel(
    const float* __restrict__ x,        // [B, D, S]
    const float* __restrict__ W,        // [D, S]
    const float* __restrict__ soma_w,   // [D]
    float* __restrict__ dst)            // ws [B, ND_T] or out [B]
{
    // double-buffered x tile: [buf][batch_local(16)][dend_local(8)][syn(60)]
    __shared__ __align__(16) float sh[2][BT * TILE_ELEMS]; // 2 x 30720 B

    const int lane  = threadIdx.x;        // 0..31 (wave32)
    const int b0    = blockIdx.x * BT;
    const int dbase = blockIdx.y * DRANGE;

    const int koff = (lane < 16) ? 0 : 2; // K slots this lane owns in A/B operands
    const int bl   = lane & 15;           // N = batch column this lane owns in B/C

    // ---- Tensor Data Mover: one 2D-tile descriptor per chunk ----
    // Tile: 16 rows x 480 f32, row stride 245760 f32, into contiguous LDS buffer.
    // D# group1 is chunk-invariant; group0 carries per-chunk global/LDS addresses.
    v8i g1;
    g1[0] = 0x00020000;                       // data_size=2 (4B); no pad/iterate/mcast
    g1[1] = (int)((TILE_ELEMS & 0xFFFF) << 16);                // tensor_dim0[15:0]
    g1[2] = (int)(((TILE_ELEMS >> 16) & 0xFFFF) | (BT << 16)); // td0[31:16] | td1[15:0]
    g1[3] = (int)(TILE_ELEMS << 16);          // td1[31:16]=0 | tile_dim0=480
    g1[4] = BT;                               // tile_dim1=16, tile_dim2=0
    g1[5] = (int)ROW_STRIDE_E;                // tensor_dim0_stride[31:0]
    g1[6] = 0;                                // stride0[47:32]=0 | stride1[15:0]=0
    g1[7] = 0;                                // tensor_dim1_stride (unused for 2D)
    const v4i gz4 = {0, 0, 0, 0};             // groups 2/3: 2D tensor -> zero
    const v8i gz8 = {0, 0, 0, 0, 0, 0, 0, 0}; // trailing group (6-arg builtin form)

    auto issue = [&](int dc) {
        const int buf = dc & 1;
        const uint32_t lds_addr = (uint32_t)(uintptr_t)(&sh[buf][0]);
        const uint64_t gaddr = (uint64_t)(uintptr_t)x +
            ((uint64_t)((uint32_t)b0 * NDENDS + (uint32_t)(dbase + dc * DCH))) *
                (NSYNS * 4u);
        v4u g0;
        g0[0] = 1u;                                       // count=1 (valid user D#)
        g0[1] = lds_addr;                                 // lds_addr[31:0]
        g0[2] = (uint32_t)gaddr;                          // global_addr[31:0]
        g0[3] = (uint32_t)((gaddr >> 32) & 0x01FFFFFFu)   // global_addr[56:32]
                | 0x80000000u;                            // type=2 ("image")
        asm volatile("" ::: "memory");
        __builtin_amdgcn_tensor_load_to_lds(g0, g1, gz4, gz4, gz8, 0);
    };

    v8f c = {0.f, 0.f, 0.f, 0.f, 0.f, 0.f, 0.f, 0.f};

    issue(0);
    for (int dc = 0; dc < NCHUNK; ++dc) {
        if (dc + 1 < NCHUNK) {
            issue(dc + 1);
            // allow only the freshly issued TDM op outstanding -> chunk dc is in LDS
            __builtin_amdgcn_s_wait_tensorcnt((short)1);
        } else {
            __builtin_amdgcn_s_wait_tensorcnt((short)0);
        }
        asm volatile("" ::: "memory");
        const int buf = dc & 1;

        // Per-lane: 4 dot-60s for dendrite locals {koff, koff+1, 4+koff, 4+koff+1},
        // batch = b0+bl. These land exactly in this lane's B-operand slots.
        float act[4];
        #pragma unroll
        for (int j = 0; j < 2; ++j) {
            #pragma unroll
            for (int r = 0; r < 2; ++r) {
                const int ddl = 4 * j + koff + r;
                const int gd  = dbase + dc * DCH + ddl;     // global dendrite index
                const float4* wrow = (const float4*)(W + (size_t)gd * NSYNS);
                const float4* xrow = (const float4*)(&sh[buf][(bl * DCH + ddl) * NSYNS]);
                float acc = 0.f;
                #pragma unroll
                for (int i = 0; i < 15; ++i) {
                    const float4 a = xrow[i];
                    const float4 b = wrow[i];
                    acc = fmaf(a.x, b.x, acc);
                    acc = fmaf(a.y, b.y, acc);
                    acc = fmaf(a.z, b.z, acc);
                    acc = fmaf(a.w, b.w, acc);
                }
                const float rr = acc > 0.f ? acc : 0.f;
                act[2 * j + r] = (gd < CUT) ? (1.0f / (1.0f + expf(-rr)))
                                            : sqrtf(rr);
            }
        }

        // Fused soma reduction via WMMA:
        //   A[m,k] = soma_w[dbase + dc*8 + 4j + k]  (identical over m -> broadcast)
        //   B[k,n] = act(dendrite 4j+k, batch b0+n)
        //   C[m,n] += sum_k soma_w[k] * act[k, n]   (same value in every row m)
        #pragma unroll
        for (int j = 0; j < 2; ++j) {
            const int gk = dbase + dc * DCH + 4 * j + koff;
            v2f A; A.x = soma_w[gk];     A.y = soma_w[gk + 1];
            v2f B; B.x = act[2 * j];     B.y = act[2 * j + 1];
            c = __builtin_amdgcn_wmma_f32_16x16x4_f32(
                    false, A, false, B, (short)0, c, false, false);
        }
    }

    // C rows are identical; lane l (l<16) holds column N=l -> partial for batch b0+l.
    if (lane < 16) {
        if (USE_WS) {
            dst[(size_t)(b0 + lane) * ND_T + blockIdx.y] = c[0];
        } else {
            atomicAdd(&dst[b0 + lane], c[0]);
        }
    }
}

__global__ void soma_reduce_kernel(const float* __restrict__ ws,
                                   const float* __restrict__ soma_b,
                                   float* __restrict__ out)
{
    const int b = blockIdx.x * blockDim.x + threadIdx.x;
    if (b < NBATCH) {
        const float* p = ws + (size_t)b * ND_T;
        float s = 0.f;
        #pragma unroll 8
        for (int t = 0; t < ND_T; ++t) s += p[t];
        out[b] = s + soma_b[0];
    }
}

__global__ void init_out_kernel(float* __restrict__ out,
                                const float* __restrict__ soma_b)
{
    const int b = blockIdx.x * blockDim.x + threadIdx.x;
    if (b < NBATCH) out[b] = soma_b[0];
}

extern "C" void kernel_launch(void* const* d_in, const int* in_sizes, int n_in,
                              void* d_out, int out_size, void* d_ws, size_t ws_size,
                              hipStream_t stream)
{
    (void)in_sizes; (void)n_in; (void)out_size;
    const float* x      = (const float*)d_in[0];
    const float* W      = (const float*)d_in[1];
    const float* soma_w = (const float*)d_in[2];
    const float* soma_b = (const float*)d_in[3];
    float* out = (float*)d_out;

    const size_t ws_needed = (size_t)NBATCH * ND_T * sizeof(float); // 256 KB
    dim3 grid(NB_T, ND_T);

    if (ws_size >= ws_needed && d_ws != nullptr) {
        float* ws = (float*)d_ws;
        dend_stream_kernel<true><<<grid, 32, 0, stream>>>(x, W, soma_w, ws);
        soma_reduce_kernel<<<(NBATCH + 255) / 256, 256, 0, stream>>>(ws, soma_b, out);
    } else {
        init_out_kernel<<<(NBATCH + 255) / 256, 256, 0, stream>>>(out, soma_b);
        dend_stream_kernel<false><<<grid, 32, 0, stream>>>(x, W, soma_w, out);
    }
}